// Decoder_33835752358310
// MI455X (gfx1250) — compile-verified
//
#include <hip/hip_runtime.h>
#include <hip/hip_bf16.h>
#include <math.h>

typedef float v2f __attribute__((ext_vector_type(2)));
typedef float v8f __attribute__((ext_vector_type(8)));

#define B_SZ   128
#define N_SZ   512
#define D_IN   256
#define D_ATT  256
#define H_NUM  8
#define DK     32
#define T_STEPS 64
#define KVL_COLS 768

// ---------------------------------------------------------------------------
// Kernel 0: fixed_ctx[b][d] = (mean_n embedding[b,n,:]) @ W_ctx
// ---------------------------------------------------------------------------
__global__ void __launch_bounds__(256) ctx_kernel(const float* __restrict__ emb,
                                                  const float* __restrict__ Wctx,
                                                  float* __restrict__ fixed_ctx) {
    const int b = blockIdx.x;
    const int d = threadIdx.x;                 // 256 threads, one per dim
    __shared__ float meanv[D_IN];
    const float* e = emb + (size_t)b * N_SZ * D_IN;
    float s = 0.f;
    for (int n = 0; n < N_SZ; ++n) s += e[(size_t)n * D_IN + d];   // coalesced
    meanv[d] = s * (1.0f / (float)N_SZ);
    __syncthreads();
    float acc = 0.f;
    for (int i = 0; i < D_IN; ++i) acc += meanv[i] * Wctx[i * D_ATT + d];
    fixed_ctx[b * D_ATT + d] = acc;
}

// ---------------------------------------------------------------------------
// Kernel 1: kvl = embedding @ W_kvl via V_WMMA_F32_16X16X4_F32.
// Block = 256 threads = 8 waves. Each wave computes TWO adjacent 16x16 N-tiles
// from one A load (register blocking). The 32-column W_kvl slab is staged in
// LDS as two stride-16 arrays (keeps the lane0-15 / lane16-31 K-row split on
// disjoint bank groups -> conflict-free ds reads).
// Segment routing (gK/gV/lK) and batch index are BLOCK-uniform, so the
// epilogue is branch-free: one base pointer + constant stride per tile.
// Grid = (65536/16/8) m-supertiles * (768/32) n-blocks = 512*24 = 12288.
// ---------------------------------------------------------------------------
__global__ void __launch_bounds__(256) kvl_gemm_kernel(const float* __restrict__ emb,
                                                       const float* __restrict__ Wkvl,
                                                       float* __restrict__ gK,
                                                       float* __restrict__ gV,
                                                       float* __restrict__ lK) {
    const int n_block = blockIdx.x % (KVL_COLS / 32);   // 0..23
    const int m_super = blockIdx.x / (KVL_COLS / 32);
    const int wave = threadIdx.x >> 5;
    const int lane = threadIdx.x & 31;
    const int m_tile = m_super * 8 + wave;

    __shared__ float Bs0[D_IN * 16];   // cols [n_block*32,      +16)
    __shared__ float Bs1[D_IN * 16];   // cols [n_block*32 + 16, +16)
    for (int i = threadIdx.x; i < D_IN * 16; i += 256) {
        const int k = i >> 4, c = i & 15;
        const float* wrow = Wkvl + (size_t)k * KVL_COLS + n_block * 32;
        Bs0[i] = wrow[c];
        Bs1[i] = wrow[16 + c];
    }
    __syncthreads();

    const int row = lane & 15;          // A row (M) == B col (N) lane slot
    const int kh  = (lane >> 4) << 1;   // K sub-offset: 0 or 2
    const float* A = emb + ((size_t)m_tile * 16 + row) * D_IN;

    v8f c0 = {0.f, 0.f, 0.f, 0.f, 0.f, 0.f, 0.f, 0.f};
    v8f c1 = {0.f, 0.f, 0.f, 0.f, 0.f, 0.f, 0.f, 0.f};
    #pragma unroll 4
    for (int k0 = 0; k0 < D_IN; k0 += 4) {
        float2 av = *reinterpret_cast<const float2*>(A + k0 + kh);   // 8B aligned
        v2f a; a.x = av.x; a.y = av.y;
        v2f b0; b0.x = Bs0[(k0 + kh) * 16 + row]; b0.y = Bs0[(k0 + kh + 1) * 16 + row];
        v2f b1; b1.x = Bs1[(k0 + kh) * 16 + row]; b1.y = Bs1[(k0 + kh + 1) * 16 + row];
        c0 = __builtin_amdgcn_wmma_f32_16x16x4_f32(false, a, false, b0,
                                                   (short)0, c0, false, false);
        c1 = __builtin_amdgcn_wmma_f32_16x16x4_f32(false, a, false, b1,
                                                   (short)0, c1, false, false);
    }

    // ---- branch-free epilogue (segment + batch are uniform for the tile)
    const int bIdx  = m_tile >> 5;                 // batch element (tile never crosses)
    const int n0    = (m_tile * 16) & 511;         // node row base within batch
    const int rbase = (lane >> 4) << 3;            // C layout: M = rbase + v
    const int nrow  = n0 + rbase;

    #pragma unroll
    for (int sub = 0; sub < 2; ++sub) {
        const v8f& c  = sub ? c1 : c0;
        const int colg = n_block * 32 + sub * 16 + row;   // global kvl column
        float* dst;
        size_t stride;
        if (n_block < 8) {                                // gK segment
            const int h = colg >> 5, i = colg & 31;
            dst = gK + ((((size_t)bIdx * H_NUM + h) * N_SZ) + nrow) * DK + i;
            stride = DK;
        } else if (n_block < 16) {                        // gV segment
            const int cc = colg - 256;
            const int h = cc >> 5, i = cc & 31;
            dst = gV + ((((size_t)bIdx * H_NUM + h) * N_SZ) + nrow) * DK + i;
            stride = DK;
        } else {                                          // logit_K segment
            dst = lK + ((size_t)bIdx * N_SZ + nrow) * D_ATT + (colg - 512);
            stride = D_ATT;
        }
        #pragma unroll
        for (int v = 0; v < 8; ++v) dst[(size_t)v * stride] = c[v];
    }
}

// ---------------------------------------------------------------------------
// Kernel 2: persistent greedy decode. One block per batch element; 256 threads
// (8 wave32s). Runs all 64 steps with no global synchronization.
// ---------------------------------------------------------------------------
__global__ void __launch_bounds__(256) decode_kernel(const float* __restrict__ emb,
                                                     const float* __restrict__ Wstep,
                                                     const float* __restrict__ Wout,
                                                     const float* __restrict__ fixed_ctx,
                                                     const float* __restrict__ gK,
                                                     const float* __restrict__ gV,
                                                     const float* __restrict__ lK,
                                                     float* __restrict__ logp_out,
                                                     float* __restrict__ pi_out) {
    const int b    = blockIdx.x;
    const int tid  = threadIdx.x;
    const int lane = tid & 31;
    const int wave = tid >> 5;

    __shared__ float q[D_ATT];
    __shared__ float nodeE[D_IN];
    __shared__ float attn[H_NUM * N_SZ];      // scores -> attn weights (16KB)
    __shared__ float heads[D_ATT];
    __shared__ float glimpse[D_ATT];
    __shared__ float logits[N_SZ];
    __shared__ float redf[8];
    __shared__ int   redi[8];
    __shared__ float sMax, sLse;
    __shared__ int   sCur;
    __shared__ unsigned visited[N_SZ / 32];   // 512-bit mask

    if (tid == 0) sCur = 0;
    if (tid < N_SZ / 32) visited[tid] = 0u;
    __syncthreads();

    const float* embB = emb + (size_t)b * N_SZ * D_IN;
    const float* gKb  = gK  + (size_t)b * H_NUM * N_SZ * DK;
    const float* gVb  = gV  + (size_t)b * H_NUM * N_SZ * DK;
    const float* lKb  = lK  + (size_t)b * N_SZ * D_ATT;
    const float* fc   = fixed_ctx + b * D_ATT;
    const float inv_sqrt_dk = 0.17677669529663687f;   // 1/sqrt(32)
    const float inv_sqrt_d  = 0.0625f;                // 1/sqrt(256)

    for (int t = 0; t < T_STEPS; ++t) {
        const int cur = sCur;
        // ---- stage current node embedding (coalesced, shared by all threads)
        nodeE[tid] = embB[(size_t)cur * D_IN + tid];
        __syncthreads();
        // ---- query[d] = fixed_ctx + node_emb @ W_step + W_step[256]+W_step[257]
        {
            float acc = fc[tid] + Wstep[256 * D_ATT + tid] + Wstep[257 * D_ATT + tid];
            for (int i = 0; i < D_IN; ++i) acc += nodeE[i] * Wstep[i * D_ATT + tid];
            q[tid] = acc;
        }
        __syncthreads();

        // ---- per-head attention: wave h owns head h
        {
            const int h = wave;
            const float* Kh = gKb + (size_t)h * N_SZ * DK;
            for (int n = lane; n < N_SZ; n += 32) {
                const float4* kp = reinterpret_cast<const float4*>(Kh + (size_t)n * DK);
                float s = 0.f;
                #pragma unroll
                for (int j = 0; j < DK / 4; ++j) {
                    const float4 kv = kp[j];
                    s += kv.x * q[h * DK + 4 * j + 0] + kv.y * q[h * DK + 4 * j + 1]
                       + kv.z * q[h * DK + 4 * j + 2] + kv.w * q[h * DK + 4 * j + 3];
                }
                const bool vis = (visited[n >> 5] >> (n & 31)) & 1u;
                attn[h * N_SZ + n] = vis ? -INFINITY : s * inv_sqrt_dk;
            }
            // wave softmax over 512 entries
            float m = -INFINITY;
            for (int n = lane; n < N_SZ; n += 32) m = fmaxf(m, attn[h * N_SZ + n]);
            for (int off = 16; off; off >>= 1) m = fmaxf(m, __shfl_xor(m, off, 32));
            float ssum = 0.f;
            for (int n = lane; n < N_SZ; n += 32) {
                const float v = attn[h * N_SZ + n];
                const float e = (v == -INFINITY) ? 0.f : __expf(v - m);
                attn[h * N_SZ + n] = e;
                ssum += e;
            }
            for (int off = 16; off; off >>= 1) ssum += __shfl_xor(ssum, off, 32);
            const float inv = 1.0f / ssum;
            // heads: lane = dk index, coalesced 128B reads of gV
            const float* Vh = gVb + (size_t)h * N_SZ * DK;
            float acc = 0.f;
            for (int n = 0; n < N_SZ; ++n)
                acc += attn[h * N_SZ + n] * Vh[(size_t)n * DK + lane];
            heads[h * DK + lane] = acc * inv;
        }
        __syncthreads();

        // ---- glimpse = heads @ W_out
        {
            float acc = 0.f;
            for (int i = 0; i < D_ATT; ++i) acc += heads[i] * Wout[i * D_ATT + tid];
            glimpse[tid] = acc;
        }
        __syncthreads();

        // ---- logits[n] = tanh(glimpse . logit_K[n] / sqrt(D)) * 10, masked
        __builtin_prefetch((const void*)(lKb + (size_t)(tid + 256) * D_ATT), 0, 1);
        for (int n = tid; n < N_SZ; n += 256) {
            const float4* kp = reinterpret_cast<const float4*>(lKb + (size_t)n * D_ATT);
            float s = 0.f;
            #pragma unroll 8
            for (int j = 0; j < D_ATT / 4; ++j) {
                const float4 kv = kp[j];
                s += kv.x * glimpse[4 * j] + kv.y * glimpse[4 * j + 1]
                   + kv.z * glimpse[4 * j + 2] + kv.w * glimpse[4 * j + 3];
            }
            s = tanhf(s * inv_sqrt_d) * 10.0f;
            const bool vis = (visited[n >> 5] >> (n & 31)) & 1u;
            logits[n] = vis ? -INFINITY : s;
        }
        __syncthreads();

        // ---- block max
        {
            float m = fmaxf(logits[tid], logits[tid + 256]);
            for (int off = 16; off; off >>= 1) m = fmaxf(m, __shfl_xor(m, off, 32));
            if (lane == 0) redf[wave] = m;
            __syncthreads();
            if (tid == 0) {
                float mm = redf[0];
                for (int w = 1; w < 8; ++w) mm = fmaxf(mm, redf[w]);
                sMax = mm;
            }
            __syncthreads();
        }
        const float m = sMax;
        // ---- block sum of exp
        {
            float s = 0.f;
            for (int n = tid; n < N_SZ; n += 256) {
                const float v = logits[n];
                if (v != -INFINITY) s += __expf(v - m);
            }
            for (int off = 16; off; off >>= 1) s += __shfl_xor(s, off, 32);
            if (lane == 0) redf[wave] = s;
            __syncthreads();
            if (tid == 0) {
                float ss = 0.f;
                for (int w = 0; w < 8; ++w) ss += redf[w];
                sLse = __logf(ss);
            }
            __syncthreads();
        }
        const float lse = sLse;

        // ---- emit log_p row and first-index argmax
        {
            int cand = 0x7fffffff;
            float* outRow = logp_out + ((size_t)b * T_STEPS + t) * N_SZ;
            for (int n = tid; n < N_SZ; n += 256) {
                const float v = logits[n];
                outRow[n] = (v == -INFINITY) ? -INFINITY : (v - m - lse);
                if (v == m && n < cand) cand = n;   // first max index
            }
            for (int off = 16; off; off >>= 1) cand = min(cand, __shfl_xor(cand, off, 32));
            if (lane == 0) redi[wave] = cand;
            __syncthreads();
            if (tid == 0) {
                int sel = redi[0];
                for (int w = 1; w < 8; ++w) sel = min(sel, redi[w]);
                sCur = sel;
                visited[sel >> 5] |= (1u << (sel & 31));
                pi_out[b * T_STEPS + t] = (float)sel;
            }
            __syncthreads();
        }
    }
}

// ---------------------------------------------------------------------------
extern "C" void kernel_launch(void* const* d_in, const int* in_sizes, int n_in,
                              void* d_out, int out_size, void* d_ws, size_t ws_size,
                              hipStream_t stream) {
    const float* emb   = (const float*)d_in[0];   // [128,512,256]
    const float* Wctx  = (const float*)d_in[1];   // [256,256]
    const float* Wkvl  = (const float*)d_in[2];   // [256,768]
    const float* Wstep = (const float*)d_in[3];   // [258,256]
    const float* Wout  = (const float*)d_in[4];   // [256,256]

    float* ws = (float*)d_ws;
    float* fixed_ctx = ws;                                     //  32768 floats
    float* gK = fixed_ctx + B_SZ * D_ATT;                      //  16.78M floats
    float* gV = gK + (size_t)B_SZ * H_NUM * N_SZ * DK;
    float* lK = gV + (size_t)B_SZ * H_NUM * N_SZ * DK;

    float* logp = (float*)d_out;                               // [128,64,512]
    float* pi   = logp + (size_t)B_SZ * T_STEPS * N_SZ;        // [128,64] as float

    ctx_kernel<<<B_SZ, 256, 0, stream>>>(emb, Wctx, fixed_ctx);

    const int mSupers = (B_SZ * N_SZ) / 16 / 8;   // 512
    const int nBlocks = KVL_COLS / 32;            // 24
    kvl_gemm_kernel<<<mSupers * nBlocks, 256, 0, stream>>>(emb, Wkvl, gK, gV, lK);

    decode_kernel<<<B_SZ, 256, 0, stream>>>(emb, Wstep, Wout, fixed_ctx,
                                            gK, gV, lK, logp, pi);
}